// DPR_23124103922233
// MI455X (gfx1250) — compile-verified
//
#include <hip/hip_runtime.h>
#include <cmath>

// Problem constants (from reference): B=512 queries, P=1024 passages,
// M=16 tokens, D=768 hidden. n=256, top-k=102, bottom-l=204.
#define BQ 512
#define PP 1024
#define MT 16
#define DD 768

typedef float v2f __attribute__((ext_vector_type(2)));
typedef float v8f __attribute__((ext_vector_type(8)));

__device__ __forceinline__ float softplus_f(float x) {
  return (x > 20.0f) ? x : log1pf(expf(x));
}

#define KC 48               // K-chunk (768 = 16 chunks)
#define NCH (DD / KC)       // 16
#define LDSTR (KC + 4)      // padded LDS row stride (floats): 13*m mod 16 permutation
                            // -> conflict-free ds_load_b64 fragment reads

// CDNA5 direct global->LDS async copy (ASYNCcnt-tracked), with fallback.
#if __has_builtin(__builtin_amdgcn_global_load_async_to_lds_b128) && \
    __has_builtin(__builtin_amdgcn_s_wait_asynccnt)
#define USE_ASYNC_LDS 1
#else
#define USE_ASYNC_LDS 0
#endif

#if USE_ASYNC_LDS
typedef int v4i_vs __attribute__((vector_size(16)));
typedef __attribute__((address_space(1))) v4i_vs* async_gptr_t;  // global b128
typedef __attribute__((address_space(3))) v4i_vs* async_lptr_t;  // LDS b128
#endif

__device__ __forceinline__ void lds_copy16(const float* g, float* l) {
#if USE_ASYNC_LDS
  __builtin_amdgcn_global_load_async_to_lds_b128(
      (async_gptr_t)g, (async_lptr_t)l, /*imm offset=*/0, /*cpol=*/0);
#else
  *reinterpret_cast<float4*>(l) = *reinterpret_cast<const float4*>(g);
#endif
}

__device__ __forceinline__ void wait_stage_done() {
#if USE_ASYNC_LDS
  __builtin_amdgcn_s_wait_asynccnt(0);
#endif
}

// One workgroup = 8 waves = query b x 8 passages. Each wave computes one
// 16x16 score tile with V_WMMA_F32_16X16X4_F32 (exact f32), then
// bitonic-sorts the 256 scores fully in registers and emits
// logit = alpha*(S - prefix154) - beta*prefix204.
__global__ __launch_bounds__(256) void dpr_logits_kernel(
    const float* __restrict__ q_emb, const float* __restrict__ p_emb,
    const float* __restrict__ alpha_raw, const float* __restrict__ beta_raw,
    float* __restrict__ out)  // out[0] = loss (written later), out+1 = logits [B,P]
{
  __shared__ float qs[2][MT * LDSTR];       // double-buffered 16 x 48 f32
  __shared__ float ps[2][8][MT * LDSTR];    // double-buffered 8 x 16 x 48 f32

  const int tid  = threadIdx.x;
  const int lane = tid & 31;
  const int wave = tid >> 5;                // 0..7 -> passage within block
  const int b    = blockIdx.y;
  const int pblk = blockIdx.x;
  const int p    = pblk * 8 + wave;

  const int half = lane >> 4;               // 0/1 -> K offset {0,2} per ISA A/B layout
  const int mrow = lane & 15;               // M for A, N for B (B = p^T -> p row)

  // ---- staging: issue copies of one K-chunk into one LDS buffer ----
  // q tile: 16x48 = 192 float4 (threads 0..191); p tiles: 8x16x48 = 1536 float4 (6/thread)
  auto stage_chunk = [&](int k0, int buf) {
    if (tid < 192) {
      const int row = tid / 12;
      const int col = (tid % 12) << 2;
      lds_copy16(q_emb + ((size_t)b * MT + row) * DD + k0 + col,
                 &qs[buf][row * LDSTR + col]);
    }
#pragma unroll
    for (int i = 0; i < 6; ++i) {
      const int idx  = tid + i * 256;       // 0..1535
      const int pass = idx / 192;
      const int rem  = idx % 192;
      const int row  = rem / 12;
      const int col  = (rem % 12) << 2;
      lds_copy16(p_emb + ((size_t)(pblk * 8 + pass) * MT + row) * DD + k0 + col,
                 &ps[buf][pass][row * LDSTR + col]);
    }
  };

  v8f c = {0.f, 0.f, 0.f, 0.f, 0.f, 0.f, 0.f, 0.f};

  stage_chunk(0, 0);  // prologue: chunk 0 -> buffer 0

  for (int ci = 0; ci < NCH; ++ci) {
    const int cur = ci & 1;
    wait_stage_done();        // my async copies for chunk ci have landed in LDS
    __syncthreads();          // all waves' chunk-ci data visible; buf[1-cur] free

    if (ci + 1 < NCH)         // prefetch next chunk, overlapped with WMMAs below
      stage_chunk((ci + 1) * KC, 1 - cur);

    const float* qb = qs[cur];
    const float* pw = ps[cur][wave];
    // 12 WMMA steps per chunk: one ds b64 fragment load each for A and B.
#pragma unroll
    for (int kk = 0; kk < KC; kk += 4) {
      const int koff = kk + (half << 1);
      v2f a, bb;
      a.x  = qb[mrow * LDSTR + koff];
      a.y  = qb[mrow * LDSTR + koff + 1];
      bb.x = pw[mrow * LDSTR + koff];       // B[k][n] = p[n][k] -> contiguous row
      bb.y = pw[mrow * LDSTR + koff + 1];
      c = __builtin_amdgcn_wmma_f32_16x16x4_f32(
          /*neg_a=*/false, a, /*neg_b=*/false, bb,
          /*c_mod=*/(short)0, c, /*reuse_a=*/false, /*reuse_b=*/false);
    }
  }

  // ---- In-register bitonic sort of the 256 scores (element e = lane*8 + r) ----
  float v[8];
#pragma unroll
  for (int r = 0; r < 8; ++r) v[r] = c[r];

#pragma unroll
  for (int k = 2; k <= 256; k <<= 1) {
#pragma unroll
    for (int j = k >> 1; j > 0; j >>= 1) {
      if (j >= 8) {                          // cross-lane stride: lane ^= (j/8)
        const int lmask = j >> 3;
#pragma unroll
        for (int r = 0; r < 8; ++r) {
          const float o = __shfl_xor(v[r], lmask, 32);
          const int e = lane * 8 + r;
          const bool keepmin = (((e & k) == 0) == ((e & j) == 0));
          const float mn = fminf(v[r], o);
          const float mx = fmaxf(v[r], o);
          v[r] = keepmin ? mn : mx;
        }
      } else {                               // in-lane stride between registers
#pragma unroll
        for (int r = 0; r < 8; ++r) {
          if ((r & j) == 0) {
            const int r2 = r | j;
            const int e = lane * 8 + r;
            const bool asc = ((e & k) == 0);
            const float mn = fminf(v[r], v[r2]);
            const float mx = fmaxf(v[r], v[r2]);
            v[r]  = asc ? mn : mx;
            v[r2] = asc ? mx : mn;
          }
        }
      }
    }
  }

  // top_sum = S - prefix(154); bottom_sum = prefix(204)  (k=102, l=204, n=256)
  float s_all = 0.f, s154 = 0.f, s204 = 0.f;
#pragma unroll
  for (int r = 0; r < 8; ++r) {
    const int e = lane * 8 + r;
    s_all += v[r];
    if (e < 154) s154 += v[r];
    if (e < 204) s204 += v[r];
  }
#pragma unroll
  for (int m = 16; m > 0; m >>= 1) {
    s_all += __shfl_xor(s_all, m, 32);
    s154  += __shfl_xor(s154, m, 32);
    s204  += __shfl_xor(s204, m, 32);
  }

  if (lane == 0) {
    const float alpha = softplus_f(alpha_raw[0]);
    const float beta  = softplus_f(beta_raw[0]);
    const float logit = alpha * (s_all - s154) - beta * s204;
    out[1 + (size_t)b * PP + p] = logit;
  }
}

// Per-row NLL contribution: lse(row) - row[b]
__global__ __launch_bounds__(256) void row_loss_kernel(
    const float* __restrict__ logits, float* __restrict__ rowloss)
{
  __shared__ float red[256];
  const int b = blockIdx.x;
  const int tid = threadIdx.x;
  const float* row = logits + (size_t)b * PP;

  float mx = -3.402823466e+38f;
  for (int i = tid; i < PP; i += 256) mx = fmaxf(mx, row[i]);
  red[tid] = mx; __syncthreads();
  for (int s = 128; s > 0; s >>= 1) {
    if (tid < s) red[tid] = fmaxf(red[tid], red[tid + s]);
    __syncthreads();
  }
  mx = red[0];
  __syncthreads();

  float se = 0.f;
  for (int i = tid; i < PP; i += 256) se += expf(row[i] - mx);
  red[tid] = se; __syncthreads();
  for (int s = 128; s > 0; s >>= 1) {
    if (tid < s) red[tid] += red[tid + s];
    __syncthreads();
  }
  if (tid == 0) rowloss[b] = (mx + logf(red[0])) - row[b];
}

__global__ __launch_bounds__(256) void final_loss_kernel(
    const float* __restrict__ rowloss, float* __restrict__ out)
{
  __shared__ float red[256];
  const int tid = threadIdx.x;
  float s = 0.f;
  for (int i = tid; i < BQ; i += 256) s += rowloss[i];
  red[tid] = s; __syncthreads();
  for (int st = 128; st > 0; st >>= 1) {
    if (tid < st) red[tid] += red[tid + st];
    __syncthreads();
  }
  if (tid == 0) out[0] = red[0] / (float)BQ;
}

extern "C" void kernel_launch(void* const* d_in, const int* in_sizes, int n_in,
                              void* d_out, int out_size, void* d_ws, size_t ws_size,
                              hipStream_t stream) {
  const float* q  = (const float*)d_in[0];  // [512,16,768] f32
  const float* pe = (const float*)d_in[1];  // [1024,16,768] f32
  const float* ar = (const float*)d_in[2];  // alpha_raw [1]
  const float* br = (const float*)d_in[3];  // beta_raw  [1]
  float* out = (float*)d_out;               // [1 + 512*1024]: loss, logits
  float* rowws = (float*)d_ws;              // >= 512 floats scratch

  dim3 grid(PP / 8, BQ);
  dpr_logits_kernel<<<grid, 256, 0, stream>>>(q, pe, ar, br, out);
  row_loss_kernel<<<BQ, 256, 0, stream>>>(out + 1, rowws);
  final_loss_kernel<<<1, 256, 0, stream>>>(rowws, out);
}